// Selfmix_40742059770566
// MI455X (gfx1250) — compile-verified
//
#include <hip/hip_runtime.h>
#include <math.h>

typedef float v2f __attribute__((ext_vector_type(2)));
typedef float v8f __attribute__((ext_vector_type(8)));

// ---------------- constant coupling tables ----------------
// COUPLINGS in reference order (lout-major, then l1, then l2)
__device__ __constant__ int CQ_LO[19]   = {0,0,0,1,1,1,1,1,1,2,2,2,2,2,2,3,3,3,4};
__device__ __constant__ int CQ_L1[19]   = {0,1,2,0,1,1,1,2,2,0,1,1,2,2,2,1,2,2,2};
__device__ __constant__ int CQ_L2[19]   = {0,1,2,1,0,1,2,1,2,2,1,2,0,1,2,2,1,2,2};
__device__ __constant__ int CQ_BASE[20] = {0,1,10,35,44,53,80,125,170,245,270,315,390,415,490,615,720,825,1000,1225};
__device__ __constant__ int CQ_MOFF[19] = {0,128,256,384,512,640,768,896,1024,1152,1280,1408,1536,1664,1792,1920,2048,2176,2304};
__device__ __constant__ int COLOFF[5]   = {0,1,4,9,16};

// device globals: CG table (1225 used) and per-channel scaled B matrices [c][68 K-rows][32 cols]
__device__ float g_cgtab[1232];
__device__ float g_Bmat[128][68 * 32];

// ---------------- stage 1: real-basis Clebsch-Gordan on device ----------------
__device__ __forceinline__ double dfact(int n) {
  double r = 1.0;
  for (int i = 2; i <= n; ++i) r *= (double)i;
  return r;
}

__device__ double cg_complex(int l1, int l2, int l3, int m1, int m2) {
  int m3 = m1 + m2;
  if (m3 < -l3 || m3 > l3) return 0.0;
  double pre = sqrt((2.0 * l3 + 1.0) * dfact(l1 + l2 - l3) * dfact(l1 - l2 + l3) *
                    dfact(-l1 + l2 + l3) / dfact(l1 + l2 + l3 + 1));
  pre *= sqrt(dfact(l3 + m3) * dfact(l3 - m3) * dfact(l1 - m1) * dfact(l1 + m1) *
              dfact(l2 - m2) * dfact(l2 + m2));
  double s = 0.0;
  for (int k = 0; k <= l1 + l2 - l3; ++k) {
    int d0 = k, d1 = l1 + l2 - l3 - k, d2 = l1 - m1 - k, d3 = l2 + m2 - k;
    int d4 = l3 - l2 + m1 + k, d5 = l3 - l1 - m2 + k;
    if (d0 < 0 || d1 < 0 || d2 < 0 || d3 < 0 || d4 < 0 || d5 < 0) continue;
    double den = dfact(d0) * dfact(d1) * dfact(d2) * dfact(d3) * dfact(d4) * dfact(d5);
    s += ((k & 1) ? -1.0 : 1.0) / den;
  }
  return pre * s;
}

// row a of complex->real basis matrix U(l); <=2 nonzero complex columns
__device__ __forceinline__ int urow(int l, int a, int* ci, double* ur, double* ui) {
  const double is2 = 0.70710678118654752440;
  int m = a - l;
  if (m == 0) { ci[0] = l; ur[0] = 1.0; ui[0] = 0.0; return 1; }
  if (m > 0) {
    ci[0] = l + m; ur[0] = (m & 1) ? -is2 : is2; ui[0] = 0.0;
    ci[1] = l - m; ur[1] = is2;                  ui[1] = 0.0;
    return 2;
  }
  int mm = -m;
  ci[0] = l - mm; ur[0] = 0.0; ui[0] = is2;
  ci[1] = l + mm; ur[1] = 0.0; ui[1] = (mm & 1) ? is2 : -is2;
  return 2;
}

__device__ float cg_real_entry(int l1, int l2, int l3, int a, int b, int k) {
  int i1[2]; double r1[2], q1[2]; int n1 = urow(l1, a, i1, r1, q1);
  int i2[2]; double r2[2], q2[2]; int n2 = urow(l2, b, i2, r2, q2);
  int i3[2]; double r3[2], q3[2]; int n3 = urow(l3, k, i3, r3, q3);
  double accr = 0.0, acci = 0.0;
  for (int p = 0; p < n1; ++p)
    for (int q = 0; q < n2; ++q)
      for (int t = 0; t < n3; ++t) {
        int m1 = i1[p] - l1, m2 = i2[q] - l2;
        if (i3[t] - l3 != m1 + m2) continue;
        double c = cg_complex(l1, l2, l3, m1, m2);
        double ar = r1[p], ai = q1[p], br = r2[q], bi = q2[q];
        double cr = r3[t], cim = -q3[t];  // conj(U3)
        double xr = ar * br - ai * bi, xi = ar * bi + ai * br;
        double yr = xr * cr - xi * cim, yi = xr * cim + xi * cr;
        accr += yr * c; acci += yi * c;
      }
  return (float)((fabs(accr) >= fabs(acci)) ? accr : acci);
}

__device__ __forceinline__ void decode_entry(int e, int& q, int& i, int& j, int& k) {
  q = 0;
  while (q < 18 && e >= CQ_BASE[q + 1]) ++q;
  int local = e - CQ_BASE[q];
  int nlo = 2 * CQ_LO[q] + 1, n2 = 2 * CQ_L2[q] + 1;
  k = local % nlo;
  j = (local / nlo) % n2;
  i = local / (nlo * n2);
}

__global__ void __launch_bounds__(256) cg_kernel() {
  int e = blockIdx.x * blockDim.x + threadIdx.x;
  if (e >= 1225) return;
  int q, i, j, k;
  decode_entry(e, q, i, j, k);
  g_cgtab[e] = cg_real_entry(CQ_L1[q], CQ_L2[q], CQ_LO[q], i, j, k);
}

// ---------------- stage 2: per-channel combined/scaled B matrix ----------------
__device__ __forceinline__ int rowmap(int l1, int l2, int i, int j) {
  if (l1 > l2) { int t = l1; l1 = l2; l2 = t; t = i; i = j; j = t; }  // x_a*x_b == x_b*x_a
  int base;
  if (l1 == 0)      base = (l2 == 0) ? 0 : ((l2 == 1) ? 1 : 4);
  else if (l1 == 1) base = (l2 == 1) ? 9 : 18;
  else              base = 33;
  return base + i * (2 * l2 + 1) + j;
}

__global__ void __launch_bounds__(256) bmat_kernel(const float* __restrict__ keep,
                                                   const float* __restrict__ mix) {
  __shared__ float Bs[68 * 32];
  const int c = blockIdx.x;      // channel
  const int tid = threadIdx.x;
  for (int e = tid; e < 68 * 32; e += 256) Bs[e] = 0.0f;
  __syncthreads();
  for (int e = tid; e < 1225; e += 256) {
    int q, i, j, k;
    decode_entry(e, q, i, j, k);
    int row = rowmap(CQ_L1[q], CQ_L2[q], i, j);
    int col = COLOFF[CQ_LO[q]] + k;
    float val = 0.5f * mix[CQ_MOFF[q] + c] * g_cgtab[e];
    atomicAdd(&Bs[row * 32 + col], val);
  }
  __syncthreads();
  if (tid < 9) {  // keep/skip path: raw-x rows 58..66, diagonal keep coefficients
    int koff = (tid == 0) ? 0 : ((tid < 4) ? 128 : 256);
    Bs[(58 + tid) * 32 + tid] = keep[koff + c];
  }
  __syncthreads();
  for (int e = tid; e < 68 * 32; e += 256) g_Bmat[c][e] = Bs[e];
}

// ---------------- stage 3: WMMA tensor-product kernel ----------------
// product table: row -> xv[PA]*xv[PB]  (PB=-1 -> raw value, PA=-1 -> zero pad)
__device__ __forceinline__ float prodc(const float* xv, int r) {
  constexpr int PA[68] = {
      0,                         // x0*x0
      0, 0, 0,                   // x0*x1
      0, 0, 0, 0, 0,             // x0*x2
      1, 1, 1, 2, 2, 2, 3, 3, 3, // x1 (x) x1
      1, 1, 1, 1, 1, 2, 2, 2, 2, 2, 3, 3, 3, 3, 3,                 // x1 (x) x2
      4, 4, 4, 4, 4, 5, 5, 5, 5, 5, 6, 6, 6, 6, 6, 7, 7, 7, 7, 7, 8, 8, 8, 8, 8, // x2 (x) x2
      0, 1, 2, 3, 4, 5, 6, 7, 8, // keep rows (raw x)
      -1};
  constexpr int PB[68] = {
      0,
      1, 2, 3,
      4, 5, 6, 7, 8,
      1, 2, 3, 1, 2, 3, 1, 2, 3,
      4, 5, 6, 7, 8, 4, 5, 6, 7, 8, 4, 5, 6, 7, 8,
      4, 5, 6, 7, 8, 4, 5, 6, 7, 8, 4, 5, 6, 7, 8, 4, 5, 6, 7, 8, 4, 5, 6, 7, 8,
      -1, -1, -1, -1, -1, -1, -1, -1, -1,
      -1};
  int a = PA[r], b = PB[r];
  if (a < 0) return 0.0f;
  float v = xv[a];
  return (b >= 0) ? v * xv[b] : v;
}

__global__ void __launch_bounds__(256) tp_kernel(const float* __restrict__ x,
                                                 float* __restrict__ out) {
  const int c    = blockIdx.x;          // channel 0..127
  const int wave = threadIdx.x >> 5;    // 8 waves per block
  const int lane = threadIdx.x & 31;
  const int half = lane >> 4;           // K-half select (A/B operand layout)
  const int n    = lane & 15;           // M row (A) / N column (B,C,D)

  // load this channel's B matrix into registers once, reused across tiles
  const float* bm = &g_Bmat[c][0];
  v2f B0[17], B1[7];
#pragma unroll
  for (int s = 0; s < 17; ++s) {
    int r0 = 4 * s + 2 * half;
    B0[s].x = bm[r0 * 32 + n];
    B0[s].y = bm[(r0 + 1) * 32 + n];
  }
#pragma unroll
  for (int s = 0; s < 7; ++s) {
    int r0 = 4 * (s + 8) + 2 * half;     // K rows 32..59 feed columns 16..24
    B1[s].x = bm[r0 * 32 + 16 + n];
    B1[s].y = bm[(r0 + 1) * 32 + 16 + n];
  }

  const int nt = gridDim.y * 8;
  for (int t = blockIdx.y * 8 + wave; t < 1024; t += nt) {
    const int b = t * 16 + n;
    const float* xp = x + (size_t)b * 1152 + c;
    float xv[9];
#pragma unroll
    for (int i = 0; i < 9; ++i) xv[i] = xp[i * 128];

    v8f acc0 = {0.f, 0.f, 0.f, 0.f, 0.f, 0.f, 0.f, 0.f};
    v8f acc1 = {0.f, 0.f, 0.f, 0.f, 0.f, 0.f, 0.f, 0.f};

#pragma unroll
    for (int s = 0; s < 17; ++s) {
      float p0 = prodc(xv, 4 * s + 0), p1 = prodc(xv, 4 * s + 1);
      float p2 = prodc(xv, 4 * s + 2), p3 = prodc(xv, 4 * s + 3);
      v2f a;
      a.x = half ? p2 : p0;
      a.y = half ? p3 : p1;
      acc0 = __builtin_amdgcn_wmma_f32_16x16x4_f32(false, a, false, B0[s],
                                                   (short)0, acc0, false, false);
    }
#pragma unroll
    for (int s = 0; s < 7; ++s) {
      int r0 = 32 + 4 * s;
      float p0 = prodc(xv, r0 + 0), p1 = prodc(xv, r0 + 1);
      float p2 = prodc(xv, r0 + 2), p3 = prodc(xv, r0 + 3);
      v2f a;
      a.x = half ? p2 : p0;
      a.y = half ? p3 : p1;
      acc1 = __builtin_amdgcn_wmma_f32_16x16x4_f32(false, a, false, B1[s],
                                                   (short)0, acc1, false, false);
    }

    // D layout: lane n holds column n; VGPR r holds batch row r (+8 for upper lane half)
    float* op = out + (size_t)(t * 16) * 3200 + c;
#pragma unroll
    for (int r = 0; r < 8; ++r) {
      int m = r + 8 * half;
      op[(size_t)m * 3200 + n * 128] = acc0[r];
    }
    if (n < 9) {
#pragma unroll
      for (int r = 0; r < 8; ++r) {
        int m = r + 8 * half;
        op[(size_t)m * 3200 + (16 + n) * 128] = acc1[r];
      }
    }
  }
}

// ---------------- launch ----------------
extern "C" void kernel_launch(void* const* d_in, const int* in_sizes, int n_in,
                              void* d_out, int out_size, void* d_ws, size_t ws_size,
                              hipStream_t stream) {
  (void)in_sizes; (void)n_in; (void)out_size; (void)d_ws; (void)ws_size;
  const float* x    = (const float*)d_in[0];
  const float* keep = (const float*)d_in[1];
  const float* mix  = (const float*)d_in[2];
  float* out = (float*)d_out;

  cg_kernel<<<dim3(5), dim3(256), 0, stream>>>();
  bmat_kernel<<<dim3(128), dim3(256), 0, stream>>>(keep, mix);
  tp_kernel<<<dim3(128, 8), dim3(256), 0, stream>>>(x, out);
}